// GAT_63840393887761
// MI455X (gfx1250) — compile-verified
//
#include <hip/hip_runtime.h>
#include <hip/hip_bf16.h>

#define N_NODES 50000
#define N_EDGES 1600000
#define IN_FEAT 128
#define OUT_FEAT 32
#define N_HEADS 4
#define HF (N_HEADS * OUT_FEAT)   // 128
#define NEG_SLOPE 0.2f

typedef __attribute__((ext_vector_type(2))) float v2f;
typedef __attribute__((ext_vector_type(8))) float v8f;

// ---------------------------------------------------------------------------
// Kernel 0: initialize workspace regions that are accumulated into.
// rst <- 0, denom <- 0, m <- -inf
// ---------------------------------------------------------------------------
__global__ void gat_init_kernel(float* __restrict__ rst,
                                float* __restrict__ denom,
                                float* __restrict__ m) {
    int t = blockIdx.x * blockDim.x + threadIdx.x;
    if (t < N_NODES * HF) rst[t] = 0.0f;
    if (t < N_NODES * N_HEADS) {
        denom[t] = 0.0f;
        m[t] = -__builtin_inff();
    }
}

// ---------------------------------------------------------------------------
// Kernel 1: feat = h @ W  (M=50000, K=128, N=128), fp32 WMMA.
// One wave32 computes one 16x16 tile; K marched in steps of 4 via
// v_wmma_f32_16x16x4_f32. Block = 128 threads = 4 waves covering 4 N-tiles.
// Grid = (50000/16, 128/(16*4)) = (3125, 2).
// ---------------------------------------------------------------------------
__global__ void gat_gemm_wmma_kernel(const float* __restrict__ h,
                                     const float* __restrict__ W,
                                     float* __restrict__ feat) {
    const int lane = threadIdx.x & 31;
    const int wave = threadIdx.x >> 5;          // 0..3
    const int half = lane >> 4;                 // 0 or 1
    const int l    = lane & 15;                 // 0..15

    const int m0 = blockIdx.x * 16;                   // row tile base
    const int n0 = (blockIdx.y * 4 + wave) * 16;      // col tile base

    v8f acc = {};

    const float* hrow = h + (size_t)(m0 + l) * IN_FEAT;

    #pragma unroll 4
    for (int kk = 0; kk < IN_FEAT; kk += 4) {
        const int kb = kk + half * 2;
        // A: 16x4 tile of h. lane<16: K=kb,kb+1 for M=l ; lane>=16: K=kb(=kk+2),kb+1
        v2f a;
        a.x = hrow[kb + 0];
        a.y = hrow[kb + 1];
        // B: 4x16 tile of W (row-major K x N). Same K split across lane halves.
        v2f b;
        b.x = W[(size_t)(kb + 0) * HF + n0 + l];
        b.y = W[(size_t)(kb + 1) * HF + n0 + l];
        acc = __builtin_amdgcn_wmma_f32_16x16x4_f32(
            /*neg_a=*/false, a, /*neg_b=*/false, b,
            /*c_mod=*/(short)0, acc, /*reuse_a=*/false, /*reuse_b=*/false);
    }

    // D layout: VGPR j -> M = m0 + half*8 + j, N = n0 + l
    #pragma unroll
    for (int j = 0; j < 8; ++j) {
        feat[(size_t)(m0 + half * 8 + j) * HF + n0 + l] = acc[j];
    }
}

// ---------------------------------------------------------------------------
// Kernel 2: el[n,h] = dot(feat[n,h,:], attn_l[h,:]) ; er likewise.
// One thread per (node, head).
// ---------------------------------------------------------------------------
__global__ void gat_attn_proj_kernel(const float* __restrict__ feat,
                                     const float* __restrict__ attn_l,
                                     const float* __restrict__ attn_r,
                                     float* __restrict__ el,
                                     float* __restrict__ er) {
    int t = blockIdx.x * blockDim.x + threadIdx.x;
    if (t >= N_NODES * N_HEADS) return;
    const int n  = t >> 2;
    const int hd = t & 3;
    const float* fp = feat + (size_t)n * HF + hd * OUT_FEAT;
    const float* al = attn_l + hd * OUT_FEAT;
    const float* ar = attn_r + hd * OUT_FEAT;
    float sl = 0.0f, sr = 0.0f;
    #pragma unroll
    for (int f = 0; f < OUT_FEAT; ++f) {
        const float v = fp[f];
        sl += v * al[f];
        sr += v * ar[f];
    }
    el[t] = sl;
    er[t] = sr;
}

// ---------------------------------------------------------------------------
// Float atomic max via sign-split int/uint atomics (valid for mixed signs,
// init = -inf). Nonneg floats order as ints; negatives order reversed as uints.
// ---------------------------------------------------------------------------
__device__ __forceinline__ void atomicMaxFloat(float* addr, float v) {
    if (v >= 0.0f) {
        atomicMax((int*)addr, __float_as_int(v));
    } else {
        atomicMin((unsigned int*)addr, __float_as_uint(v));
    }
}

// ---------------------------------------------------------------------------
// Kernel 3: e[edge,h] = leaky_relu(el[src]+er[dst]); segment max into m[dst,h].
// ---------------------------------------------------------------------------
__global__ void gat_edge_score_kernel(const float* __restrict__ el,
                                      const float* __restrict__ er,
                                      const int* __restrict__ src,
                                      const int* __restrict__ dst,
                                      float* __restrict__ ebuf,
                                      float* __restrict__ m) {
    int t = blockIdx.x * blockDim.x + threadIdx.x;
    if (t >= N_EDGES * N_HEADS) return;
    const int eidx = t >> 2;
    const int hd   = t & 3;
    const int s = src[eidx];
    const int d = dst[eidx];
    float v = el[s * N_HEADS + hd] + er[d * N_HEADS + hd];
    v = (v > 0.0f) ? v : v * NEG_SLOPE;
    ebuf[t] = v;
    atomicMaxFloat(&m[d * N_HEADS + hd], v);
}

// ---------------------------------------------------------------------------
// Kernel 4: zero-in-degree guard: m == -inf -> 0
// ---------------------------------------------------------------------------
__global__ void gat_mfix_kernel(float* __restrict__ m) {
    int t = blockIdx.x * blockDim.x + threadIdx.x;
    if (t >= N_NODES * N_HEADS) return;
    const float v = m[t];
    if (!(v == v) || v == -__builtin_inff() || v == __builtin_inff()) m[t] = 0.0f;
}

// ---------------------------------------------------------------------------
// Kernel 5: ex = exp(e - m[dst]); in-place into ebuf; denom[dst] += ex.
// ---------------------------------------------------------------------------
__global__ void gat_edge_exp_kernel(const int* __restrict__ dst,
                                    const float* __restrict__ m,
                                    float* __restrict__ ebuf,
                                    float* __restrict__ denom) {
    int t = blockIdx.x * blockDim.x + threadIdx.x;
    if (t >= N_EDGES * N_HEADS) return;
    const int eidx = t >> 2;
    const int hd   = t & 3;
    const int d = dst[eidx];
    const float ex = __expf(ebuf[t] - m[d * N_HEADS + hd]);
    ebuf[t] = ex;
    atomicAdd(&denom[d * N_HEADS + hd], ex);
}

// ---------------------------------------------------------------------------
// Kernel 6: heavy scatter. One 32-lane group per edge; lane j-loop covers the
// 128 features (head = j, feat = lane). rst[dst] += feat[src] * alpha.
// Consecutive lanes hit consecutive addresses -> coalesced cacheline atomics.
// ---------------------------------------------------------------------------
__global__ void gat_aggregate_kernel(const float* __restrict__ feat,
                                     const float* __restrict__ ebuf,
                                     const float* __restrict__ denom,
                                     const int* __restrict__ src,
                                     const int* __restrict__ dst,
                                     float* __restrict__ rst) {
    int t = blockIdx.x * blockDim.x + threadIdx.x;
    const int eidx = t >> 5;
    const int lane = t & 31;
    if (eidx >= N_EDGES) return;
    const int s = src[eidx];
    const int d = dst[eidx];
    const float* fs = feat + (size_t)s * HF;
    float* rd = rst + (size_t)d * HF;
    #pragma unroll
    for (int hd = 0; hd < N_HEADS; ++hd) {
        const float dn = denom[d * N_HEADS + hd];
        const float alpha = ebuf[eidx * N_HEADS + hd] / fmaxf(dn, 1e-9f);
        const int c = hd * OUT_FEAT + lane;
        atomicAdd(&rd[c], fs[c] * alpha);
    }
}

// ---------------------------------------------------------------------------
// Kernel 7: out[n,f] = mean_h relu(rst[n,h,f] + bias[h,f])
// ---------------------------------------------------------------------------
__global__ void gat_epilogue_kernel(const float* __restrict__ rst,
                                    const float* __restrict__ bias,
                                    float* __restrict__ out) {
    int t = blockIdx.x * blockDim.x + threadIdx.x;
    if (t >= N_NODES * OUT_FEAT) return;
    const int n = t >> 5;
    const int f = t & 31;
    float sum = 0.0f;
    #pragma unroll
    for (int hd = 0; hd < N_HEADS; ++hd) {
        const int c = hd * OUT_FEAT + f;
        sum += fmaxf(rst[(size_t)n * HF + c] + bias[c], 0.0f);
    }
    out[t] = sum * (1.0f / N_HEADS);
}

// ---------------------------------------------------------------------------
// Launcher
// ---------------------------------------------------------------------------
extern "C" void kernel_launch(void* const* d_in, const int* in_sizes, int n_in,
                              void* d_out, int out_size, void* d_ws, size_t ws_size,
                              hipStream_t stream) {
    const float* h      = (const float*)d_in[0];
    const float* W      = (const float*)d_in[1];
    const float* attn_l = (const float*)d_in[2];
    const float* attn_r = (const float*)d_in[3];
    const float* bias   = (const float*)d_in[4];
    const int*   src    = (const int*)d_in[5];
    const int*   dst    = (const int*)d_in[6];
    float* out = (float*)d_out;

    // workspace carve-up (floats)
    float* ws    = (float*)d_ws;
    float* feat  = ws;                                // N*128
    float* rst   = feat  + (size_t)N_NODES * HF;      // N*128
    float* el    = rst   + (size_t)N_NODES * HF;      // N*4
    float* er    = el    + (size_t)N_NODES * N_HEADS; // N*4
    float* m     = er    + (size_t)N_NODES * N_HEADS; // N*4
    float* denom = m     + (size_t)N_NODES * N_HEADS; // N*4
    float* ebuf  = denom + (size_t)N_NODES * N_HEADS; // E*4

    const int BS = 256;

    // 0) init accumulators
    {
        int n = N_NODES * HF;
        gat_init_kernel<<<(n + BS - 1) / BS, BS, 0, stream>>>(rst, denom, m);
    }
    // 1) feat = h @ W via fp32 WMMA
    {
        dim3 grid(N_NODES / 16, HF / (16 * 4));   // (3125, 2)
        gat_gemm_wmma_kernel<<<grid, 128, 0, stream>>>(h, W, feat);
    }
    // 2) el / er
    {
        int n = N_NODES * N_HEADS;
        gat_attn_proj_kernel<<<(n + BS - 1) / BS, BS, 0, stream>>>(feat, attn_l, attn_r, el, er);
    }
    // 3) edge scores + segment max
    {
        int n = N_EDGES * N_HEADS;
        gat_edge_score_kernel<<<(n + BS - 1) / BS, BS, 0, stream>>>(el, er, src, dst, ebuf, m);
    }
    // 4) -inf guard
    {
        int n = N_NODES * N_HEADS;
        gat_mfix_kernel<<<(n + BS - 1) / BS, BS, 0, stream>>>(m);
    }
    // 5) exp + segment sum
    {
        int n = N_EDGES * N_HEADS;
        gat_edge_exp_kernel<<<(n + BS - 1) / BS, BS, 0, stream>>>(dst, m, ebuf, denom);
    }
    // 6) weighted scatter aggregation (dominant phase)
    {
        long long n = (long long)N_EDGES * 32;
        gat_aggregate_kernel<<<(unsigned)((n + BS - 1) / BS), BS, 0, stream>>>(
            feat, ebuf, denom, src, dst, rst);
    }
    // 7) bias + relu + head mean
    {
        int n = N_NODES * OUT_FEAT;
        gat_epilogue_kernel<<<(n + BS - 1) / BS, BS, 0, stream>>>(rst, bias, out);
    }
}